// GNNModel_32134945308761
// MI455X (gfx1250) — compile-verified
//
#include <hip/hip_runtime.h>
#include <hip/hip_bf16.h>

typedef __attribute__((ext_vector_type(2))) float v2f;
typedef __attribute__((ext_vector_type(8))) float v8f;

#define HDIM 128
#define NNODES 20000
#define NEDGES 640000
#define NGRAPH 256

// ---------------------------------------------------------------------------
// z[i] = h[i]   (self term of GIN aggregation, float4 vectorized)
// ---------------------------------------------------------------------------
__global__ void copy_f4_kernel(const float4* __restrict__ src,
                               float4* __restrict__ dst, int n4) {
    int i = blockIdx.x * blockDim.x + threadIdx.x;
    if (i < n4) dst[i] = src[i];
}

// ---------------------------------------------------------------------------
// Edge scatter-add: z[dst[e]] += h[src[e]].  One wave (32 lanes) per edge,
// each lane moves a float4 (128 floats / edge) and issues 4 f32 atomics.
// Atomics land in L2 (dest buffer is 10MB << 192MB L2).
// ---------------------------------------------------------------------------
__global__ void scatter_add_kernel(const float* __restrict__ H,
                                   const int* __restrict__ esrc,
                                   const int* __restrict__ edst,
                                   float* __restrict__ Z, int nedges) {
    int e = blockIdx.x * (blockDim.x >> 5) + (threadIdx.x >> 5);
    if (e >= nedges) return;
    int lane = threadIdx.x & 31;
    int s = esrc[e];
    int d = edst[e];
    const float4 v = *(const float4*)(H + (size_t)s * HDIM + lane * 4);
    float* out = Z + (size_t)d * HDIM + lane * 4;
    atomicAdd(out + 0, v.x);
    atomicAdd(out + 1, v.y);
    atomicAdd(out + 2, v.z);
    atomicAdd(out + 3, v.w);
}

// ---------------------------------------------------------------------------
// Out[nrows x 128] = act(Z[nrows x 128] @ W[128 x 128] + bias)
// One 16x16 output tile per wave via V_WMMA_F32_16X16X4_F32 (K-loop of 32).
// Block = 256 threads = 8 waves; wave w owns column tile w*16; block owns 16
// rows. nrows (20000) is an exact multiple of 16.
//
// Fragment layouts (ISA 7.12.2, 32-bit, wave32):
//   A 16x4 : lane m=lane&15, half h=lane>>4 -> VGPR0 = A[m][2h], VGPR1 = A[m][2h+1]
//   B 4x16 : lane n=lane&15, half h        -> VGPR0 = B[2h][n], VGPR1 = B[2h+1][n]
//   C 16x16: VGPR r -> row r (lanes 0-15) / row r+8 (lanes 16-31), col = lane&15
// ---------------------------------------------------------------------------
__global__ void gemm_bias_act_wmma(const float* __restrict__ Z,
                                   const float* __restrict__ W,
                                   const float* __restrict__ bias,
                                   float* __restrict__ Out,
                                   int nrows, int do_relu) {
    const int wave = threadIdx.x >> 5;
    const int lane = threadIdx.x & 31;
    const int row0 = blockIdx.x * 16;
    const int col0 = wave * 16;
    if (row0 >= nrows) return;

    const int m  = lane & 15;        // row within tile (A) / col within tile (B,C)
    const int kh = (lane >> 4) << 1; // 0 or 2 : K sub-offset for this half-wave

    v8f c = {};
    const float* zrow = Z + (size_t)(row0 + m) * HDIM;
    const float* wcol = W + col0 + m;

    #pragma unroll 4
    for (int k = 0; k < HDIM; k += 4) {
        v2f a, b;
        a.x = zrow[k + kh];
        a.y = zrow[k + kh + 1];
        b.x = wcol[(size_t)(k + kh) * HDIM];
        b.y = wcol[(size_t)(k + kh + 1) * HDIM];
        // (neg_a, A, neg_b, B, c_mod, C, reuse_a, reuse_b)
        c = __builtin_amdgcn_wmma_f32_16x16x4_f32(
                false, a, false, b, (short)0, c, false, false);
    }

    // Epilogue: bias + optional ReLU, scatter C layout back to row-major.
    const int n     = lane & 15;
    const int mbase = (lane >> 4) << 3; // 0 or 8
    const float bc  = bias[col0 + n];
    #pragma unroll
    for (int r = 0; r < 8; ++r) {
        float v = c[r] + bc;
        if (do_relu) v = fmaxf(v, 0.0f);
        Out[(size_t)(row0 + mbase + r) * HDIM + col0 + n] = v;
    }
}

// ---------------------------------------------------------------------------
// Pooling accumulation: sums[batch[n]] += h[n]; counts[batch[n]] += 1
// One wave per node, float4 per lane.
// ---------------------------------------------------------------------------
__global__ void pool_accum_kernel(const float* __restrict__ H,
                                  const int* __restrict__ batch,
                                  float* __restrict__ sums,
                                  float* __restrict__ counts, int nnodes) {
    int node = blockIdx.x * (blockDim.x >> 5) + (threadIdx.x >> 5);
    if (node >= nnodes) return;
    int lane = threadIdx.x & 31;
    int g = batch[node];
    const float4 v = *(const float4*)(H + (size_t)node * HDIM + lane * 4);
    float* out = sums + (size_t)g * HDIM + lane * 4;
    atomicAdd(out + 0, v.x);
    atomicAdd(out + 1, v.y);
    atomicAdd(out + 2, v.z);
    atomicAdd(out + 3, v.w);
    if (lane == 0) atomicAdd(counts + g, 1.0f);
}

// ---------------------------------------------------------------------------
// Head: pooled = sums/max(counts,1); out = relu(pooled@W1+b1) @ W2 + b2
// One block (128 threads) per graph; tiny GEMV, LDS-staged.
// ---------------------------------------------------------------------------
__global__ void head_kernel(const float* __restrict__ sums,
                            const float* __restrict__ counts,
                            const float* __restrict__ w1,   // [128 x 64]
                            const float* __restrict__ b1,   // [64]
                            const float* __restrict__ w2,   // [64]
                            const float* __restrict__ b2,   // [1]
                            float* __restrict__ out) {
    __shared__ float pooled[HDIM];
    __shared__ float hid[64];
    const int g = blockIdx.x;
    const int t = threadIdx.x;
    const float cnt = fmaxf(counts[g], 1.0f);
    pooled[t] = sums[(size_t)g * HDIM + t] / cnt;
    __syncthreads();
    if (t < 64) {
        float acc = b1[t];
        #pragma unroll 8
        for (int k = 0; k < HDIM; ++k) acc += pooled[k] * w1[k * 64 + t];
        hid[t] = fmaxf(acc, 0.0f) * w2[t];
    }
    __syncthreads();
    if (t == 0) {
        float acc = b2[0];
        #pragma unroll
        for (int k = 0; k < 64; ++k) acc += hid[k];
        out[g] = acc;
    }
}

// ---------------------------------------------------------------------------
extern "C" void kernel_launch(void* const* d_in, const int* in_sizes, int n_in,
                              void* d_out, int out_size, void* d_ws, size_t ws_size,
                              hipStream_t stream) {
    const float* x     = (const float*)d_in[0];
    const int*   eidx  = (const int*)d_in[1];      // [2, E] row-major
    const int*   batch = (const int*)d_in[2];
    const int    E     = in_sizes[1] / 2;
    const int*   esrc  = eidx;
    const int*   edst  = eidx + E;

    const float* w1[3] = {(const float*)d_in[3], (const float*)d_in[7],  (const float*)d_in[11]};
    const float* b1[3] = {(const float*)d_in[4], (const float*)d_in[8],  (const float*)d_in[12]};
    const float* w2[3] = {(const float*)d_in[5], (const float*)d_in[9],  (const float*)d_in[13]};
    const float* b2[3] = {(const float*)d_in[6], (const float*)d_in[10], (const float*)d_in[14]};
    const float* hw1 = (const float*)d_in[15];
    const float* hb1 = (const float*)d_in[16];
    const float* hw2 = (const float*)d_in[17];
    const float* hb2 = (const float*)d_in[18];

    // workspace layout (floats)
    const size_t feat = (size_t)NNODES * HDIM;   // 2,560,000 floats
    float* bufA  = (float*)d_ws;
    float* bufB  = bufA + feat;
    float* bufC  = bufB + feat;
    float* sums  = bufC + feat;                  // [G, 128]
    float* cnts  = sums + (size_t)NGRAPH * HDIM; // [G]

    const int n4        = (int)(feat / 4);
    const int copyGrid  = (n4 + 255) / 256;
    const int edgeGrid  = (E + 7) / 8;           // 8 edges (waves) per 256-thread block
    const int gemmGrid  = NNODES / 16;           // 20000 % 16 == 0
    const int poolGrid  = (NNODES + 7) / 8;

    // buffer rotation: h -> z -> t -> h_out
    const float* hptr[3] = {x,    bufC, bufA};
    float*       zptr[3] = {bufA, bufA, bufC};
    float*       optr[3] = {bufC, bufA, bufC};

    for (int l = 0; l < 3; ++l) {
        copy_f4_kernel<<<copyGrid, 256, 0, stream>>>(
            (const float4*)hptr[l], (float4*)zptr[l], n4);
        scatter_add_kernel<<<edgeGrid, 256, 0, stream>>>(
            hptr[l], esrc, edst, zptr[l], E);
        gemm_bias_act_wmma<<<gemmGrid, 256, 0, stream>>>(
            zptr[l], w1[l], b1[l], bufB, NNODES, 1);
        gemm_bias_act_wmma<<<gemmGrid, 256, 0, stream>>>(
            bufB, w2[l], b2[l], optr[l], NNODES, 1);
    }

    hipMemsetAsync(sums, 0, ((size_t)NGRAPH * HDIM + NGRAPH) * sizeof(float), stream);
    pool_accum_kernel<<<poolGrid, 256, 0, stream>>>(optr[2], batch, sums, cnts, NNODES);
    head_kernel<<<NGRAPH, HDIM, 0, stream>>>(sums, cnts, hw1, hb1, hw2, hb2, (float*)d_out);
}